// CrfLayer_36876589203577
// MI455X (gfx1250) — compile-verified
//
#include <hip/hip_runtime.h>
#include <math.h>

// CRF: B=256 batches, S=512 timesteps, T=128 tags.
#define CB 256
#define CS 512
#define CT 128
#define MTILE 16     // batches per block (M dim of WMMA tiles)
#define QSTR 132     // padded LDS row stride (floats): 132 mod 64 = 4 -> conflict-free
#define RSTR 132     // 132*4 = 528 bytes, 16B aligned -> b128 LDS ops legal

typedef __attribute__((ext_vector_type(2))) float v2f;
typedef __attribute__((ext_vector_type(8))) float v8f;

// ---------------------------------------------------------------------------
// CDNA5 async global->LDS copy (ASYNCcnt-tracked, no VGPR staging).
// GVS mode: mem = SGPR64 base + VGPR32 byte offset. LDS dest = per-lane VGPR
// holding the LDS byte address (low 32 bits of the generic pointer).
// ---------------------------------------------------------------------------
__device__ __forceinline__ void async_ld_f4_to_lds(const float* gbase,
                                                   unsigned gByteOff,
                                                   void* ldsPtr) {
    unsigned ldsAddr = (unsigned)(uintptr_t)ldsPtr;
    asm volatile("global_load_async_to_lds_b128 %0, %1, %2"
                 :
                 : "v"(ldsAddr), "v"(gByteOff), "s"(gbase)
                 : "memory");
}
__device__ __forceinline__ void wait_async0() {
    asm volatile("s_wait_asynccnt 0x0" ::: "memory");
}

// ---------------------------------------------------------------------------
// Kernel 1: gold path score per batch.
// ---------------------------------------------------------------------------
__global__ __launch_bounds__(128) void crf_gold_score(
    const float* __restrict__ emissions,
    const int*   __restrict__ tags,
    const float* __restrict__ mask,
    const float* __restrict__ transitions,
    const float* __restrict__ start_transitions,
    const float* __restrict__ end_transitions,
    float* __restrict__ gold)
{
    __shared__ float ssum[128];
    __shared__ float smsk[128];
    const int b   = blockIdx.x;
    const int tid = threadIdx.x;
    const int*   tg = tags + (size_t)b * CS;
    const float* mk = mask + (size_t)b * CS;
    const float* em = emissions + (size_t)b * CS * CT;

    float acc = 0.f, msum = 0.f;
    for (int s = tid; s < CS; s += 128) {
        float mm = mk[s];
        msum += mm;
        if (s >= 1) {
            int prev = tg[s - 1], cur = tg[s];
            acc += (transitions[prev * CT + cur] + em[(size_t)s * CT + cur]) * mm;
        }
    }
    ssum[tid] = acc;
    smsk[tid] = msum;
    __syncthreads();
    for (int off = 64; off > 0; off >>= 1) {
        if (tid < off) { ssum[tid] += ssum[tid + off]; smsk[tid] += smsk[tid + off]; }
        __syncthreads();
    }
    if (tid == 0) {
        int t0 = tg[0];
        float sc = ssum[0] + start_transitions[t0] + em[t0];
        int last_real = (int)(smsk[0] + 0.5f) - 1;
        int lr = last_real < 0 ? 0 : last_real;
        sc += (last_real >= 0) ? end_transitions[tg[lr]] : 0.f;
        gold[b] = sc;
    }
}

// ---------------------------------------------------------------------------
// Kernel 2: forward-algorithm log normalizer via fp32 WMMA.
// logsumexp_i(s_i + T_ij) = M + log( exp(s-M) x exp(T) )  -> GEMM via
// V_WMMA_F32_16X16X4_F32.  16 batches/block, 8 waves, wave w owns N-cols
// [16w,16w+16).  3 barriers/step; emissions stream via async b128->LDS.
// ---------------------------------------------------------------------------
__global__ __launch_bounds__(256) void crf_forward_logz(
    const float* __restrict__ emissions,
    const float* __restrict__ mask,
    const float* __restrict__ transitions,
    const float* __restrict__ start_transitions,
    const float* __restrict__ end_transitions,
    float* __restrict__ logz)
{
    __shared__ float Qt[CT * QSTR];          // Qt[j*QSTR+k] = exp(trans[k][j]) (B operand, transposed)
    __shared__ float scoreS[MTILE * RSTR];   // running log-scores
    __shared__ float Pm[MTILE * RSTR];       // exp(score - rowmax)  (A operand)
    __shared__ float emS[2][MTILE * RSTR];   // double-buffered emission tiles (async dest)
    __shared__ float red[MTILE * 16];        // loop: [m*8+wave] partial maxes; final: 16x16
    __shared__ float Mrow[MTILE];
    __shared__ float maskCur[MTILE];
    __shared__ float endS[CT];

    const int tid    = threadIdx.x;
    const int lane   = tid & 31;
    const int wave   = tid >> 5;        // 0..7
    const int half   = lane >> 4;       // 0/1
    const int lane16 = lane & 15;
    const int col    = wave * 16 + lane16;
    const int tileBase = blockIdx.x * MTILE;

    // --- preamble ---
    for (int idx = tid; idx < CT * CT; idx += 256) {
        int k = idx >> 7, j = idx & 127;
        Qt[j * QSTR + k] = __expf(transitions[idx]);
    }
    for (int idx = tid; idx < CT; idx += 256) endS[idx] = end_transitions[idx];
    for (int idx = tid; idx < MTILE * CT; idx += 256) {
        int m = idx >> 7, j = idx & 127;
        scoreS[m * RSTR + j] = start_transitions[j]
                             + emissions[((size_t)(tileBase + m) * CS + 0) * CT + j];
    }
    // async-load emissions tile for t=1 into emS[1] (16B granules)
#pragma unroll
    for (int e = 0; e < 2; ++e) {
        int idx4 = tid + e * 256;            // 512 float4 = 16x128 floats
        int m = idx4 >> 5, j4 = idx4 & 31;
        unsigned gOff = (unsigned)(((((size_t)(tileBase + m) * CS + 1) * CT) + j4 * 4) * 4);
        async_ld_f4_to_lds(emissions, gOff, &emS[1][m * RSTR + j4 * 4]);
    }
    __syncthreads();
    // prime per-(row,wavegroup) partial maxes in step layout red[m*8+w]
    if (tid < 128) {
        int m = tid >> 3, w = tid & 7;
        float mx = -INFINITY;
#pragma unroll
        for (int i = 0; i < 16; ++i) mx = fmaxf(mx, scoreS[m * RSTR + w * 16 + i]);
        red[m * 8 + w] = mx;
    }
    wait_async0();
    __syncthreads();

    for (int t = 1; t < CS; ++t) {
        const int buf = t & 1;

        // (a) async prefetch of emissions[t+1] -> emS[buf^1], overlaps whole step
        if (t + 1 < CS) {
#pragma unroll
            for (int e = 0; e < 2; ++e) {
                int idx4 = tid + e * 256;
                int m = idx4 >> 5, j4 = idx4 & 31;
                unsigned gOff = (unsigned)(((((size_t)(tileBase + m) * CS + (t + 1)) * CT) + j4 * 4) * 4);
                async_ld_f4_to_lds(emissions, gOff, &emS[buf ^ 1][m * RSTR + j4 * 4]);
            }
        }

        // (b) finish row max (merge 8 wave-partials) + fetch this step's mask
        if (tid < MTILE) {
            float mx = red[tid * 8];
#pragma unroll
            for (int w = 1; w < 8; ++w) mx = fmaxf(mx, red[tid * 8 + w]);
            Mrow[tid] = mx;
            maskCur[tid] = mask[(size_t)(tileBase + tid) * CS + t];
        }
        __syncthreads();

        // (c) P = exp(score - rowmax)
        for (int idx = tid; idx < MTILE * CT; idx += 256) {
            int m = idx >> 7, j = idx & 127;
            Pm[m * RSTR + j] = __expf(scoreS[m * RSTR + j] - Mrow[m]);
        }
        __syncthreads();

        // (d) C(16x16) = P(16x128) x Q(128x16): 32 chained fp32 WMMAs (K=4).
        // A frag: lanes 0-15 -> {K=k0,k0+1}, lanes 16-31 -> {K=k0+2,k0+3}, M=lane16.
        // B frag: same K split per lane group, N=lane16 (from transposed Qt).
        v8f c = {};
#pragma unroll
        for (int kb = 0; kb < 32; ++kb) {
            const int k0 = kb * 4 + 2 * half;
            v2f a, bq;
            a.x  = Pm[lane16 * RSTR + k0];
            a.y  = Pm[lane16 * RSTR + k0 + 1];
            bq.x = Qt[col * QSTR + k0];
            bq.y = Qt[col * QSTR + k0 + 1];
            c = __builtin_amdgcn_wmma_f32_16x16x4_f32(
                    false, a, false, bq, (short)0, c, false, false);
        }

        // (e) score_j = rowmax + log(C) + em_j (masked); fold next step's
        // row-max phase 1 into intra-wave shuffle butterflies over the 16 cols.
        // C/D layout: VGPR r holds M=r (lanes 0-15) / M=r+8 (lanes 16-31), N=lane16.
#pragma unroll
        for (int r = 0; r < 8; ++r) {
            int m = r + 8 * half;
            float nxt = Mrow[m] + __logf(c[r]) + emS[buf][m * RSTR + col];
            float old = scoreS[m * RSTR + col];
            float val = (maskCur[m] > 0.5f) ? nxt : old;
            scoreS[m * RSTR + col] = val;
            float mx = val;
            mx = fmaxf(mx, __shfl_xor(mx, 1, 32));
            mx = fmaxf(mx, __shfl_xor(mx, 2, 32));
            mx = fmaxf(mx, __shfl_xor(mx, 4, 32));
            mx = fmaxf(mx, __shfl_xor(mx, 8, 32));
            if (lane16 == 0) red[m * 8 + wave] = mx;
        }

        wait_async0();       // emS[buf^1] resident before anyone reads it next step
        __syncthreads();
    }

    // --- final: logz[b] = logsumexp_k(score[b,k] + end[k])
    {
        int r = tid >> 4, sub = tid & 15;
        float mx = -INFINITY;
#pragma unroll
        for (int e = 0; e < 8; ++e) {
            int k = sub * 8 + e;
            mx = fmaxf(mx, scoreS[r * RSTR + k] + endS[k]);
        }
        red[r * 16 + sub] = mx;
        __syncthreads();
        if (tid < MTILE) {
            float m2 = red[tid * 16];
#pragma unroll
            for (int e = 1; e < 16; ++e) m2 = fmaxf(m2, red[tid * 16 + e]);
            Mrow[tid] = m2;
        }
        __syncthreads();
        float sm = 0.f;
#pragma unroll
        for (int e = 0; e < 8; ++e) {
            int k = sub * 8 + e;
            sm += __expf(scoreS[r * RSTR + k] + endS[k] - Mrow[r]);
        }
        red[r * 16 + sub] = sm;
        __syncthreads();
        if (tid < MTILE) {
            float s2 = 0.f;
#pragma unroll
            for (int e = 0; e < 16; ++e) s2 += red[tid * 16 + e];
            logz[tileBase + tid] = Mrow[tid] + __logf(s2);
        }
    }
}

// ---------------------------------------------------------------------------
// Kernel 3: ll = gold - logz; out = (-mean(ll), mean(ll))
// ---------------------------------------------------------------------------
__global__ __launch_bounds__(256) void crf_final(
    const float* __restrict__ gold,
    const float* __restrict__ logz,
    float* __restrict__ out)
{
    __shared__ float s[256];
    int tid = threadIdx.x;
    s[tid] = gold[tid] - logz[tid];
    __syncthreads();
    for (int off = 128; off > 0; off >>= 1) {
        if (tid < off) s[tid] += s[tid + off];
        __syncthreads();
    }
    if (tid == 0) {
        float mean = s[0] * (1.0f / CB);
        out[0] = -mean;
        out[1] = mean;
    }
}

extern "C" void kernel_launch(void* const* d_in, const int* in_sizes, int n_in,
                              void* d_out, int out_size, void* d_ws, size_t ws_size,
                              hipStream_t stream) {
    (void)in_sizes; (void)n_in; (void)out_size; (void)ws_size;
    const float* emissions = (const float*)d_in[0];
    const int*   tags      = (const int*)d_in[1];   // integer input -> const int* per harness
    const float* mask      = (const float*)d_in[2];
    const float* trans     = (const float*)d_in[3];
    const float* startT    = (const float*)d_in[4];
    const float* endT      = (const float*)d_in[5];

    float* gold = (float*)d_ws;        // [CB]
    float* logz = gold + CB;           // [CB]
    float* out  = (float*)d_out;       // [2]

    crf_gold_score<<<CB, 128, 0, stream>>>(emissions, tags, mask, trans, startT, endT, gold);
    crf_forward_logz<<<CB / MTILE, 256, 0, stream>>>(emissions, mask, trans, startT, endT, logz);
    crf_final<<<1, 256, 0, stream>>>(gold, logz, out);
}